// MHA_2705829397226
// MI455X (gfx1250) — compile-verified
//
#include <hip/hip_runtime.h>

// B=64, S=512, D=1024, N=16 heads, H=64, K=64 (downsample length)

typedef __attribute__((ext_vector_type(2))) float v2f;
typedef __attribute__((ext_vector_type(8))) float v8f;

__device__ __forceinline__ v8f wmma4(v2f a, v2f b, v8f c) {
  // V_WMMA_F32_16X16X4_F32 : D(16x16 f32) = A(16x4 f32) * B(4x16 f32) + C
  return __builtin_amdgcn_wmma_f32_16x16x4_f32(
      /*neg_a=*/false, a, /*neg_b=*/false, b,
      /*c_mod=*/(short)0, c, /*reuse_a=*/false, /*reuse_b=*/false);
}

// Compute NT adjacent 16x16 f32 tiles: C[t] += A(16 x 4*kSteps) * B(4*kSteps x 16)
// A row-major with leading dim lda, B row-major with leading dim ldb.
// Tile t uses B columns [t*colStride, t*colStride+16).
// ISA operand layouts (CDNA5 ISA §7.12.2):
//   A: lane L<16 -> row L, K {k,k+1}; lane L>=16 -> row L-16, K {k+2,k+3}
//   B: lane L<16 -> col L, rows {k,k+1}; lane L>=16 -> col L-16, rows {k+2,k+3}
template <int NT>
__device__ __forceinline__ void gemm_rowtile(const float* __restrict__ A, int lda,
                                             const float* __restrict__ B, int ldb,
                                             int colStride, int kSteps, v8f* acc) {
  const int lane = threadIdx.x & 31;
  const int hf = lane >> 4;       // half-wave select (K offset +2)
  const int l16 = lane & 15;
  const float* ap = A + l16 * lda + 2 * hf;
  const float* bp = B + 2 * hf * ldb + l16;
  for (int k = 0; k < kSteps; ++k) {
    v2f a = *(const v2f*)ap;
#pragma unroll
    for (int t = 0; t < NT; ++t) {
      v2f b;
      b.x = bp[t * colStride];
      b.y = bp[t * colStride + ldb];
      acc[t] = wmma4(a, b, acc[t]);
    }
    ap += 4;
    bp += 4 * ldb;
  }
}

// Store a 16x16 C-layout tile (VGPR r: lanes0-15 row r, lanes16-31 row r+8)
__device__ __forceinline__ void store_tile(float* C, int ldc, v8f acc) {
  const int lane = threadIdx.x & 31;
  const int hf = lane >> 4;
  const int l16 = lane & 15;
  float* p = C + (hf * 8) * ldc + l16;
#pragma unroll
  for (int r = 0; r < 8; ++r) p[r * ldc] = acc[r];
}

// ---------------------------------------------------------------------------
// Kernel 1: kd[b,k,d] = sum_s key_ds[k,s]*key[b,s,d]  (and vd likewise)
// Waves: [kv(2)][b(64)][kt(4)][cg(16)], each wave -> 16 rows x 64 cols
// ---------------------------------------------------------------------------
__global__ void __launch_bounds__(256) k_downsample(
    const float* __restrict__ key, const float* __restrict__ value,
    const float* __restrict__ key_ds, const float* __restrict__ value_ds,
    float* __restrict__ kd, float* __restrict__ vd) {
  int wave = (blockIdx.x * blockDim.x + threadIdx.x) >> 5;
  const int cg = wave & 15; wave >>= 4;   // 64-col group of D
  const int kt = wave & 3;  wave >>= 2;   // 16-row group of K
  const int b  = wave & 63; wave >>= 6;
  const int kv = wave;                    // 0 = key path, 1 = value path

  const float* dsm = kv ? value_ds : key_ds;   // [64, 512]
  const float* src = kv ? value : key;         // [B, 512, 1024]
  float* dst = kv ? vd : kd;                   // [B, 64, 1024]

  v8f acc[4] = {};
  gemm_rowtile<4>(dsm + (size_t)kt * 16 * 512, 512,
                  src + (size_t)b * 512 * 1024 + cg * 64, 1024,
                  16, 512 / 4, acc);
  float* C = dst + ((size_t)b * 64 + kt * 16) * 1024 + cg * 64;
#pragma unroll
  for (int t = 0; t < 4; ++t) store_tile(C + t * 16, 1024, acc[t]);
}

// ---------------------------------------------------------------------------
// Kernel 2: kh = kd x k_kernel, vh = vd x v_kernel
// Flat GEMM: [4096,1024] x [1024,1024] -> [4096,1024]
// Waves: [kv(2)][rt(256)][cg(16)]
// ---------------------------------------------------------------------------
__global__ void __launch_bounds__(256) k_project(
    const float* __restrict__ kd, const float* __restrict__ vd,
    const float* __restrict__ k_kernel, const float* __restrict__ v_kernel,
    float* __restrict__ kh, float* __restrict__ vh) {
  int wave = (blockIdx.x * blockDim.x + threadIdx.x) >> 5;
  const int cg = wave & 15;  wave >>= 4;
  const int rt = wave & 255; wave >>= 8;
  const int kv = wave;

  const float* A = (kv ? vd : kd) + (size_t)rt * 16 * 1024;
  const float* W = (kv ? v_kernel : k_kernel) + cg * 64;
  float* dst = (kv ? vh : kh) + (size_t)rt * 16 * 1024 + cg * 64;

  v8f acc[4] = {};
  gemm_rowtile<4>(A, 1024, W, 1024, 16, 1024 / 4, acc);
#pragma unroll
  for (int t = 0; t < 4; ++t) store_tile(dst + t * 16, 1024, acc[t]);
}

// ---------------------------------------------------------------------------
// Kernel 3: fused q-projection + scores + masked softmax + attn*V
// One wave per (b, 16-row s-tile, head n). Waves: [b(64)][n(16)][st(32)]
// ---------------------------------------------------------------------------
__global__ void __launch_bounds__(256) k_attention(
    const float* __restrict__ query, const float* __restrict__ q_kernel,
    const float* __restrict__ kh, const float* __restrict__ vh,
    float* __restrict__ out) {
  // per-wave 16x64 staging tile, row stride 68 (68 % 64banks == 4 -> no conflicts)
  __shared__ float smem[8][16 * 68];
  const int wid = threadIdx.x >> 5;

  int wave = (blockIdx.x * blockDim.x + threadIdx.x) >> 5;
  const int st = wave & 31; wave >>= 5;
  const int n  = wave & 15; wave >>= 4;
  const int b  = wave;
  const int s0 = st * 16;

  const int lane = threadIdx.x & 31;
  const int hf = lane >> 4;
  const int l16 = lane & 15;

  // --- 1) q head-slice: [16 x 64] = query[b, s0:+16, :1024] x Wq[:, n, :] ---
  v8f qacc[4] = {};
  gemm_rowtile<4>(query + ((size_t)b * 512 + s0) * 1024, 1024,
                  q_kernel + n * 64, 1024, 16, 1024 / 4, qacc);

  // stage q into LDS in plain row-major (C-layout -> A-layout conversion)
  float* lq = &smem[wid][0];
#pragma unroll
  for (int t = 0; t < 4; ++t) store_tile(lq + t * 16, 68, qacc[t]);
  // same-wave LDS accesses are kept in order (DScnt) -> no barrier needed

  // --- 2) scores[16 x 64] = q x kh[b,:,n,:]^T (WMMA K dim = h) ---
  v8f sacc[4] = {};
  {
    const float* ap = lq + l16 * 68 + 2 * hf;
    // B[K=h, N=kcol] = kh[(b*64 + kcol)*1024 + n*64 + h]; pairs contiguous in h
    const float* bl = kh + (((size_t)b * 64 + l16) * 16 + n) * 64 + 2 * hf;
    for (int h = 0; h < 64; h += 4) {
      v2f a = *(const v2f*)ap; ap += 4;
#pragma unroll
      for (int t = 0; t < 4; ++t) {
        v2f bv = *(const v2f*)(bl + (size_t)t * 16 * 1024 + h);
        sacc[t] = wmma4(a, bv, sacc[t]);
      }
    }
  }

  // --- 3) scale, truncated-causal mask (col <= global row), softmax over 64 ---
  const float scale = 0.125f;  // 1/sqrt(H=64)
#pragma unroll
  for (int r = 0; r < 8; ++r) {
    const int row = s0 + r + 8 * hf;   // global q index for this component
    float v[4];
    float m = -3.0e38f;
#pragma unroll
    for (int t = 0; t < 4; ++t) {
      const int col = t * 16 + l16;    // k index
      float x = sacc[t][r] * scale;
      x = (col <= row) ? x : -9.0e15f;
      v[t] = x;
      m = fmaxf(m, x);
    }
    // butterfly max/sum within each 16-lane half (halves hold different rows)
#pragma unroll
    for (int msk = 1; msk < 16; msk <<= 1) m = fmaxf(m, __shfl_xor(m, msk, 32));
    float sum = 0.f;
#pragma unroll
    for (int t = 0; t < 4; ++t) { float e = __expf(v[t] - m); v[t] = e; sum += e; }
#pragma unroll
    for (int msk = 1; msk < 16; msk <<= 1) sum += __shfl_xor(sum, msk, 32);
    const float inv = 1.0f / sum;
#pragma unroll
    for (int t = 0; t < 4; ++t) sacc[t][r] = v[t] * inv;
  }

  // stage attn into LDS (reuse buffer)
#pragma unroll
  for (int t = 0; t < 4; ++t) store_tile(lq + t * 16, 68, sacc[t]);

  // --- 4) out[16 x 64] = attn x vh[b,:,n,:]  (B row k stride = 16*64 = 1024) ---
  v8f oacc[4] = {};
  gemm_rowtile<4>(lq, 68, vh + ((size_t)b * 64 * 16 + n) * 64, 1024, 16, 64 / 4, oacc);

  float* C = out + ((size_t)b * 512 + s0) * 1024 + n * 64;
#pragma unroll
  for (int t = 0; t < 4; ++t) store_tile(C + t * 16, 1024, oacc[t]);
}

// ---------------------------------------------------------------------------
extern "C" void kernel_launch(void* const* d_in, const int* in_sizes, int n_in,
                              void* d_out, int out_size, void* d_ws, size_t ws_size,
                              hipStream_t stream) {
  const float* query    = (const float*)d_in[0];
  const float* key      = (const float*)d_in[1];
  const float* value    = (const float*)d_in[2];
  const float* q_kernel = (const float*)d_in[3];
  const float* k_kernel = (const float*)d_in[4];
  const float* v_kernel = (const float*)d_in[5];
  const float* key_ds   = (const float*)d_in[6];
  const float* value_ds = (const float*)d_in[7];
  float* out = (float*)d_out;

  // workspace: kd, vd, kh, vh : each 64*64*1024 f32 (16 MB) -> 64 MB total
  float* kd = (float*)d_ws;
  float* vd = kd + (size_t)64 * 64 * 1024;
  float* kh = vd + (size_t)64 * 64 * 1024;
  float* vh = kh + (size_t)64 * 64 * 1024;

  // k1: 2*64*4*16 = 8192 waves -> 1024 blocks of 8 waves
  k_downsample<<<1024, 256, 0, stream>>>(key, value, key_ds, value_ds, kd, vd);
  // k2: 2*256*16 = 8192 waves -> 1024 blocks
  k_project<<<1024, 256, 0, stream>>>(kd, vd, k_kernel, v_kernel, kh, vh);
  // k3: 64*16*32 = 32768 waves -> 4096 blocks
  k_attention<<<4096, 256, 0, stream>>>(query, q_kernel, kh, vh, out);
}